// HybGNN_44427141710208
// MI455X (gfx1250) — compile-verified
//
#include <hip/hip_runtime.h>
#include <hip/hip_bf16.h>

typedef __attribute__((ext_vector_type(2))) float v2f;
typedef __attribute__((ext_vector_type(8))) float v8f;
typedef __attribute__((ext_vector_type(4))) unsigned int v4u;
typedef __attribute__((ext_vector_type(8))) int v8i;
typedef __attribute__((ext_vector_type(4))) int v4i;

#define N_NODES 15
#define EMBED 2048
#define F1V 1024
#define F2V 512
#define N_EDGES 120
#define KW3 1920
#define NW3 30720   /* 15*2048 */
#define KSPLIT 16   /* split-K factor for the big GEMV (1920/16 = 120 rows each) */
#define KC 256      /* K-chunk staged into LDS by the TDM for the WMMA GEMMs */

/* ---- workspace layout (float offsets) ---- */
#define O_H0     0u        /* 480 -> pad 512   */
#define O_H1     512u      /* 1920 -> pad 2048 */
#define O_H      4096u     /* 16*2048 = 32768  */
#define O_DINV   36864u    /* 16               */
#define O_HW1    36896u    /* 16*1024 = 16384  */
#define O_X1     53280u    /* 16384            */
#define O_XW2    69664u    /* 16*512 = 8192    */
#define O_X2     77856u    /* 8192             */
#define O_CSUM   86048u    /* 512              */
#define O_CTX    86560u    /* 512              */
#define O_PART   87072u    /* KSPLIT*30720 = 491520 */

/* ---- Tensor Data Mover: load a [rows x tile_cols] f32 tile (row stride =
   row_stride_elems) from global into LDS at lds_off, packed row-major.
   D# layout per CDNA5 ISA ch.8 (groups 0/1; 2D tile -> groups 2/3 zero). ---- */
__device__ __forceinline__ void tdm_load_tile_to_lds(unsigned int lds_off,
                                                     const float* gptr,
                                                     unsigned int tile_cols,
                                                     unsigned int rows,
                                                     unsigned int row_stride_elems) {
    unsigned long long ga = (unsigned long long)(uintptr_t)gptr;
    v4u g0;
    g0[0] = 1u;                                         /* count=1 (valid), user mode */
    g0[1] = lds_off;                                    /* lds_addr (bytes) */
    g0[2] = (unsigned int)(ga & 0xFFFFFFFFu);           /* global_addr[31:0] */
    g0[3] = (unsigned int)((ga >> 32) & 0x1FFFFFFu)     /* global_addr[56:32] */
          | (2u << 30);                                 /* type = 2 ("image") */
    unsigned int td0 = row_stride_elems;                /* tensor_dim0 (elements) */
    unsigned int td1 = rows;                            /* tensor_dim1 */
    unsigned int st0 = row_stride_elems;                /* tensor_dim0_stride */
    v8i g1;
    g1[0] = (int)(2u << 16);                            /* data_size=2 (4 bytes); mask=0 */
    g1[1] = (int)((td0 & 0xFFFFu) << 16);               /* bits 63:48 = tensor_dim0[15:0] */
    g1[2] = (int)((td0 >> 16) | ((td1 & 0xFFFFu) << 16));
    g1[3] = (int)((td1 >> 16) | (tile_cols << 16));     /* bits 127:112 = tile_dim0 */
    g1[4] = (int)(rows & 0xFFFFu);                      /* bits 143:128 = tile_dim1; tile_dim2=0 */
    g1[5] = (int)st0;                                   /* tensor_dim0_stride[31:0] */
    g1[6] = 0;                                          /* stride[47:32]=0; dim1_stride[15:0]=0 */
    g1[7] = 0;
    v4i g2 = {0, 0, 0, 0};
    v4i g3 = {0, 0, 0, 0};
#if __clang_major__ >= 23
    v8i g4 = {0, 0, 0, 0, 0, 0, 0, 0};
    __builtin_amdgcn_tensor_load_to_lds(g0, g1, g2, g3, g4, 0);
#else
    __builtin_amdgcn_tensor_load_to_lds(g0, g1, g2, g3, 0);
#endif
}

/* ---- 1: h0 = relu(f @ W1 + b1), [15]x[15,480] ---- */
__global__ void k_embed1(const float* __restrict__ f, const float* __restrict__ W1,
                         const float* __restrict__ b1, float* __restrict__ h0) {
    int j = blockIdx.x * blockDim.x + threadIdx.x;
    if (j >= 480) return;
    float acc = b1[j];
#pragma unroll
    for (int i = 0; i < 15; ++i) acc += f[i] * W1[i * 480 + j];
    h0[j] = fmaxf(acc, 0.0f);
}

/* ---- 2: h1 = relu(h0 @ W2 + b2), [480]x[480,1920] ---- */
__global__ void k_embed2(const float* __restrict__ h0, const float* __restrict__ W2,
                         const float* __restrict__ b2, float* __restrict__ h1) {
    int j = blockIdx.x * blockDim.x + threadIdx.x;
    if (j >= 1920) return;
    float acc = b2[j];
#pragma unroll 4
    for (int i = 0; i < 480; ++i) acc += h0[i] * W2[i * 1920 + j];
    h1[j] = fmaxf(acc, 0.0f);
}

/* ---- 3a: split-K partials of h1 @ W3 (236 MB stream — the roofline term) ----
   gridDim = (30, KSPLIT), 256 thr; thread -> 4 consecutive columns (b128 loads). */
__global__ void k_gemv_partial(const float* __restrict__ h1, const float* __restrict__ W3,
                               float* __restrict__ part) {
    int cg = blockIdx.x * blockDim.x + threadIdx.x;     /* 0..7679 */
    int c  = cg * 4;
    int k0 = blockIdx.y * (KW3 / KSPLIT);               /* 120 rows per slice */
    const float4* wp = (const float4*)(W3 + (size_t)k0 * NW3 + c);
    const size_t rs4 = NW3 / 4;                         /* row stride in float4 */
    float ax = 0.f, ay = 0.f, az = 0.f, aw = 0.f;
#pragma unroll 4
    for (int k = 0; k < KW3 / KSPLIT; ++k) {
        float hv = h1[k0 + k];
        float4 w = *wp;
        ax += hv * w.x; ay += hv * w.y; az += hv * w.z; aw += hv * w.w;
        wp += rs4;
        __builtin_prefetch(wp + 2 * rs4, 0, 1);         /* global_prefetch_b8 */
    }
    float4 r; r.x = ax; r.y = ay; r.z = az; r.w = aw;
    *(float4*)(part + (size_t)blockIdx.y * NW3 + c) = r;
}

/* ---- 3b: reduce partials + b3 into padded H[16][2048] (row 15 zeroed) ---- */
__global__ void k_gemv_reduce(const float* __restrict__ part, const float* __restrict__ b3,
                              float* __restrict__ H) {
    int c = blockIdx.x * blockDim.x + threadIdx.x;      /* < 32768 */
    if (c >= NW3) { H[c] = 0.0f; return; }
    float s = b3[c];
#pragma unroll
    for (int sp = 0; sp < KSPLIT; ++sp) s += part[(size_t)sp * NW3 + c];
    H[c] = s;
}

/* ---- 4: degrees (A + I) and dinv = rsqrt(deg) ---- */
__global__ void k_degree(const int* __restrict__ dst, float* __restrict__ dinv) {
    __shared__ float deg[16];
    int t = threadIdx.x;
    if (t < 16) deg[t] = 1.0f;                          /* self loop */
    __syncthreads();
    if (t < N_EDGES) atomicAdd(&deg[dst[t]], 1.0f);
    __syncthreads();
    if (t < 16) dinv[t] = rsqrtf(deg[t]);
}

/* ---- 5: C[16,N] = A[16,K] @ B[K,N] via V_WMMA_F32_16X16X4_F32.
   A is staged into LDS in 16xKC chunks by the Tensor Data Mover (wave 0 issues
   the D#, waits TENSORcnt, workgroup barrier publishes the tile). All 4 waves
   then feed A fragments from LDS (ds loads) while streaming B from HBM.
   f32 fragment layouts (ISA 7.12.2):
     A 16x4: lane&15 = M, lane>>4 picks K pair {k,k+1}/{k+2,k+3}, 2 VGPRs.
     B  4x16: mirrored (lane&15 = N, half picks K pair).
     C/D: VGPR r -> row r + 8*(lane>>4), col = lane&15.
   Launch geometry is an exact multiple of N/16 waves -> EXEC all ones. ---- */
__global__ void k_wmma16(const float* __restrict__ A, const float* __restrict__ B,
                         float* __restrict__ C, int K, int N) {
    __shared__ float sA[16 * KC];                       /* 16 KB */
    int wave = threadIdx.x >> 5;
    int lane = threadIdx.x & 31;
    int n0 = (blockIdx.x * (blockDim.x >> 5) + wave) << 4;
    int half = lane >> 4;
    int l = lane & 15;
    const float* Bcol = B + n0 + l;
    /* flat LDS address low 32 bits == wave-relative LDS byte offset (ISA 10.2) */
    unsigned int lds_off = (unsigned int)(uintptr_t)&sA[0];
    v8f acc = {};
    for (int k0 = 0; k0 < K; k0 += KC) {
        if (wave == 0) {
            tdm_load_tile_to_lds(lds_off, A + k0, KC, 16, (unsigned int)K);
            __builtin_amdgcn_s_wait_tensorcnt(0);
        }
        __syncthreads();                                /* tile visible to all waves */
        const float* Ar = &sA[l * KC + 2 * half];
        for (int kk = 0; kk < KC; kk += 4) {
            v2f a, b;
            a[0] = Ar[kk];
            a[1] = Ar[kk + 1];
            int kg = k0 + kk + 2 * half;
            b[0] = Bcol[(size_t)kg * N];
            b[1] = Bcol[(size_t)(kg + 1) * N];
            acc = __builtin_amdgcn_wmma_f32_16x16x4_f32(
                /*neg_a=*/false, a, /*neg_b=*/false, b,
                /*c_mod=*/(short)0, acc, /*reuse_a=*/false, /*reuse_b=*/false);
        }
        __syncthreads();                                /* protect sA before next chunk */
    }
    float* Crow = C + n0 + l;
#pragma unroll
    for (int r = 0; r < 8; ++r) Crow[(size_t)(r + 8 * half) * N] = acc[r];
}

/* ---- 6: normalized adjacency aggregation (+bias, optional relu); n==15 -> 0 ---- */
__global__ void k_aggregate(const float* __restrict__ HW, const float* __restrict__ bias,
                            const int* __restrict__ src, const int* __restrict__ dst,
                            const float* __restrict__ dinv, float* __restrict__ X,
                            int N, int do_relu) {
    int f = blockIdx.x * blockDim.x + threadIdx.x;
    int n = blockIdx.y;
    if (f >= N) return;
    if (n >= N_NODES) { X[(size_t)n * N + f] = 0.0f; return; }
    float dn = dinv[n];
    float acc = dn * dn * HW[(size_t)n * N + f];        /* self loop */
    for (int e = 0; e < N_EDGES; ++e) {
        if (dst[e] == n) {
            int s = src[e];
            acc += dinv[s] * dn * HW[(size_t)s * N + f];
        }
    }
    acc += bias[f];
    if (do_relu) acc = fmaxf(acc, 0.0f);
    X[(size_t)n * N + f] = acc;
}

/* ---- 7: colsum[k] = sum_n X2[n,k] ---- */
__global__ void k_colsum(const float* __restrict__ X2, float* __restrict__ csum) {
    int k = blockIdx.x * blockDim.x + threadIdx.x;
    if (k >= F2V) return;
    float s = 0.0f;
#pragma unroll
    for (int n = 0; n < N_NODES; ++n) s += X2[n * F2V + k];
    csum[k] = s;
}

/* ---- 8: ctx[j] = tanh((1/15) * sum_k colsum[k]*attW[k,j]) ---- */
__global__ void k_ctx(const float* __restrict__ csum, const float* __restrict__ attW,
                      float* __restrict__ ctx) {
    int j = blockIdx.x * blockDim.x + threadIdx.x;
    if (j >= F2V) return;
    float s = 0.0f;
#pragma unroll 4
    for (int k = 0; k < F2V; ++k) s += csum[k] * attW[k * F2V + j];
    ctx[j] = tanhf(s * (1.0f / 15.0f));
}

/* ---- 9: scores -> rep -> logits -> loss & softmax (wave32 reductions) ---- */
__global__ void k_final(const float* __restrict__ X2, const float* __restrict__ ctx,
                        const float* __restrict__ fcW, const float* __restrict__ fcb,
                        const float* __restrict__ target, float* __restrict__ out) {
    __shared__ float s_scores[16];
    __shared__ float s_rep[F2V];
    __shared__ float s_logits[3];
    int tid = threadIdx.x;
    int w = tid >> 5, lane = tid & 31;

    if (w < N_NODES) {                                  /* scores[n] = sigmoid(x_n . ctx) */
        float p = 0.0f;
        for (int j = lane; j < F2V; j += 32) p += X2[w * F2V + j] * ctx[j];
        for (int o = 16; o; o >>= 1) p += __shfl_xor(p, o, 32);
        if (lane == 0) s_scores[w] = 1.0f / (1.0f + expf(-p));
    }
    __syncthreads();

    {                                                   /* rep[j] = sum_n X2[n,j]*scores[n] */
        float r = 0.0f;
#pragma unroll
        for (int n = 0; n < N_NODES; ++n) r += X2[n * F2V + tid] * s_scores[n];
        s_rep[tid] = r;
    }
    __syncthreads();

    if (tid < 32) {                                     /* logits = rep @ fcW + fcb */
        for (int c = 0; c < 3; ++c) {
            float p = 0.0f;
            for (int j = lane; j < F2V; j += 32) p += s_rep[j] * fcW[j * 3 + c];
            for (int o = 16; o; o >>= 1) p += __shfl_xor(p, o, 32);
            if (lane == 0) s_logits[c] = p + fcb[c];
        }
    }
    __syncthreads();

    if (tid == 0) {
        int cls = 0;
        if (target[1] > target[cls]) cls = 1;
        if (target[2] > target[cls]) cls = 2;
        float m = fmaxf(s_logits[0], fmaxf(s_logits[1], s_logits[2]));
        float e0 = expf(s_logits[0] - m), e1 = expf(s_logits[1] - m), e2 = expf(s_logits[2] - m);
        float sum = e0 + e1 + e2;
        out[0] = -(s_logits[cls] - m - logf(sum));      /* loss */
        out[1] = e0 / sum; out[2] = e1 / sum; out[3] = e2 / sum;  /* predictions */
    }
}

extern "C" void kernel_launch(void* const* d_in, const int* in_sizes, int n_in,
                              void* d_out, int out_size, void* d_ws, size_t ws_size,
                              hipStream_t stream) {
    const float* f1     = (const float*)d_in[0];
    const int*   edges  = (const int*)  d_in[1];
    const float* target = (const float*)d_in[2];
    const float* W1 = (const float*)d_in[3];  const float* b1 = (const float*)d_in[4];
    const float* W2 = (const float*)d_in[5];  const float* b2 = (const float*)d_in[6];
    const float* W3 = (const float*)d_in[7];  const float* b3 = (const float*)d_in[8];
    const float* gW1 = (const float*)d_in[9];  const float* gb1 = (const float*)d_in[10];
    const float* gW2 = (const float*)d_in[11]; const float* gb2 = (const float*)d_in[12];
    const float* attW = (const float*)d_in[13];
    const float* fcW = (const float*)d_in[14]; const float* fcb = (const float*)d_in[15];

    const int* src = edges;
    const int* dst = edges + N_EDGES;

    float* ws   = (float*)d_ws;
    float* h0   = ws + O_H0;
    float* h1   = ws + O_H1;
    float* H    = ws + O_H;
    float* dinv = ws + O_DINV;
    float* HW1  = ws + O_HW1;
    float* X1   = ws + O_X1;
    float* XW2  = ws + O_XW2;
    float* X2   = ws + O_X2;
    float* csum = ws + O_CSUM;
    float* ctx  = ws + O_CTX;
    float* part = ws + O_PART;
    float* out  = (float*)d_out;

    k_embed1<<<1, 512, 0, stream>>>(f1, W1, b1, h0);
    k_embed2<<<8, 256, 0, stream>>>(h0, W2, b2, h1);
    k_gemv_partial<<<dim3(30, KSPLIT), 256, 0, stream>>>(h1, W3, part);
    k_gemv_reduce<<<128, 256, 0, stream>>>(part, b3, H);
    k_degree<<<1, 128, 0, stream>>>(dst, dinv);
    /* GCN1: [16,2048] @ [2048,1024] -> 64 waves (16 blk x 4 waves) */
    k_wmma16<<<16, 128, 0, stream>>>(H, gW1, HW1, EMBED, F1V);
    k_aggregate<<<dim3(4, 16), 256, 0, stream>>>(HW1, gb1, src, dst, dinv, X1, F1V, 1);
    /* GCN2: [16,1024] @ [1024,512] -> 32 waves (8 blk x 4 waves) */
    k_wmma16<<<8, 128, 0, stream>>>(X1, gW2, XW2, F1V, F2V);
    k_aggregate<<<dim3(2, 16), 256, 0, stream>>>(XW2, gb2, src, dst, dinv, X2, F2V, 0);
    k_colsum<<<2, 256, 0, stream>>>(X2, csum);
    k_ctx<<<2, 256, 0, stream>>>(csum, attW, ctx);
    k_final<<<1, 512, 0, stream>>>(X2, ctx, fcW, fcb, target, out);
}